// dNamNNLoss_3839700762828
// MI455X (gfx1250) — compile-verified
//
#include <hip/hip_runtime.h>

// Problem constants (match the reference).
#define BB 128
#define NN 16384
#define HH 512
#define WW 512

constexpr long NPIX = (long)BB * HH * WW;   // 33,554,432 pixels
constexpr long NPTS = (long)BB * NN;        // 2,097,152 points
constexpr int  THREADS     = 256;           // 8 waves (wave32)
constexpr int  LOSS_BLOCKS = 2048;

typedef float v2f __attribute__((ext_vector_type(2)));
typedef float v4f __attribute__((ext_vector_type(4)));
typedef float v8f __attribute__((ext_vector_type(8)));

// ---------------------------------------------------------------------------
// Wave32 sum via V_WMMA_F32_16X16X4_F32.
// A (16x4 f32) is striped across all 32 lanes in 2 VGPRs:
//   lane L<16 holds A[L][0],A[L][1]; lane L>=16 holds A[L-16][2],A[L-16][3].
// With a[0]=acc, a[1]=0 and B=ones, D[m][n] = acc_m + acc_{m+16}.
// Lanes 0-15 hold D rows 0-7 in c[0..7]; lanes 16-31 hold rows 8-15.
// sum(c[0..7]) + shfl_xor(.,16) == full wave sum, in every lane.
// ---------------------------------------------------------------------------
__device__ __forceinline__ float wave_sum_wmma(float acc) {
  v2f a; a[0] = acc;  a[1] = 0.0f;
  v2f b; b[0] = 1.0f; b[1] = 1.0f;
  v8f c = {};
  c = __builtin_amdgcn_wmma_f32_16x16x4_f32(
      /*neg_a=*/false, a, /*neg_b=*/false, b,
      /*c_mod=*/(short)0, c, /*reuse_a=*/false, /*reuse_b=*/false);
  float s = ((c[0] + c[1]) + (c[2] + c[3])) + ((c[4] + c[5]) + (c[6] + c[7]));
  s += __shfl_xor(s, 16, 32);
  return s;
}

// Block reduction: WMMA wave sums -> LDS -> serial sum of 8 wave partials.
// Result valid in thread 0 only.
__device__ __forceinline__ float block_sum(float acc) {
  __shared__ float wsum[8];
  float s = wave_sum_wmma(acc);
  const int lane = threadIdx.x & 31;
  const int wave = threadIdx.x >> 5;
  if (lane == 0) wsum[wave] = s;
  __syncthreads();
  float r = 0.0f;
  if (threadIdx.x == 0) {
    const int nw = blockDim.x >> 5;
    for (int i = 0; i < nw; ++i) r += wsum[i];
  }
  return r;
}

// ---------------------------------------------------------------------------
// 1) Zero the image with regular (RT) stores -> lines become L2-resident
//    dirty so the atomic splats and the re-read hit L2, not HBM.
// ---------------------------------------------------------------------------
__global__ void zero_img(v4f* __restrict__ img, long n4) {
  const long stride = (long)gridDim.x * blockDim.x;
  for (long i = (long)blockIdx.x * blockDim.x + threadIdx.x; i < n4; i += stride)
    img[i] = (v4f){0.0f, 0.0f, 0.0f, 0.0f};
}

// ---------------------------------------------------------------------------
// 2) Bilinear splat: one thread per point, 4 hardware float atomics
//    (global_atomic_add_f32) that land in L2 (image < 192 MB L2).
// ---------------------------------------------------------------------------
__global__ void splat_kernel(const float* __restrict__ pts,
                             float* __restrict__ img) {
  const long i = (long)blockIdx.x * blockDim.x + threadIdx.x;
  if (i >= NPTS) return;
  const int b = (int)(i >> 14);                  // i / NN
  const float* p = pts + i * 3;
  float x = p[0], y = p[1], inten = p[2];
  x = fminf(fmaxf(x, 0.0f), (float)(WW - 1));
  y = fminf(fmaxf(y, 0.0f), (float)(HH - 1));
  const float x0f = floorf(x), y0f = floorf(y);
  const float fx = x - x0f, fy = y - y0f;
  const int x0 = (int)x0f, y0 = (int)y0f;
  const int x1 = min(x0 + 1, WW - 1);
  const int y1 = min(y0 + 1, HH - 1);
  float* base = img + (long)b * HH * WW;
  const float wx1 = fx, wx0 = 1.0f - fx;
  const float wy1 = fy, wy0 = 1.0f - fy;
  unsafeAtomicAdd(&base[y0 * WW + x0], inten * wx0 * wy0);
  unsafeAtomicAdd(&base[y0 * WW + x1], inten * wx1 * wy0);
  unsafeAtomicAdd(&base[y1 * WW + x0], inten * wx0 * wy1);
  unsafeAtomicAdd(&base[y1 * WW + x1], inten * wx1 * wy1);
}

// ---------------------------------------------------------------------------
// 3) Fused (img - tgt)^2 partial sums. Image reads hit L2; targets stream
//    from HBM with non-temporal loads so they don't evict the image.
//    Per-block partial via WMMA wave reduction.
// ---------------------------------------------------------------------------
__global__ void loss_partial(const v4f* __restrict__ img,
                             const v4f* __restrict__ tgt,
                             float* __restrict__ partials, long n4) {
  const long stride = (long)gridDim.x * blockDim.x;
  float acc = 0.0f;
  for (long i = (long)blockIdx.x * blockDim.x + threadIdx.x; i < n4; i += stride) {
    v4f a = img[i];
    v4f t = __builtin_nontemporal_load(&tgt[i]);
    v4f d = a - t;
    acc += (d[0] * d[0] + d[1] * d[1]) + (d[2] * d[2] + d[3] * d[3]);
  }
  const float r = block_sum(acc);
  if (threadIdx.x == 0) partials[blockIdx.x] = r;
}

// ---------------------------------------------------------------------------
// 4) Final deterministic reduction of the 2048 block partials, scale by 1/NPIX.
// ---------------------------------------------------------------------------
__global__ void final_reduce(const float* __restrict__ partials,
                             float* __restrict__ out, int n, float scale) {
  float acc = 0.0f;
  for (int i = threadIdx.x; i < n; i += blockDim.x) acc += partials[i];
  const float r = block_sum(acc);
  if (threadIdx.x == 0) out[0] = r * scale;
}

extern "C" void kernel_launch(void* const* d_in, const int* in_sizes, int n_in,
                              void* d_out, int out_size, void* d_ws, size_t ws_size,
                              hipStream_t stream) {
  const float* pts = (const float*)d_in[0];   // [B, N, 3] float32
  const float* tgt = (const float*)d_in[1];   // [B, H, W] float32
  float* img      = (float*)d_ws;             // B*H*W floats (128 MB)
  float* partials = img + NPIX;               // LOSS_BLOCKS floats

  const long n4 = NPIX / 4;

  zero_img<<<4096, THREADS, 0, stream>>>((v4f*)img, n4);

  const int splat_blocks = (int)((NPTS + THREADS - 1) / THREADS);
  splat_kernel<<<splat_blocks, THREADS, 0, stream>>>(pts, img);

  loss_partial<<<LOSS_BLOCKS, THREADS, 0, stream>>>(
      (const v4f*)img, (const v4f*)tgt, partials, n4);

  final_reduce<<<1, THREADS, 0, stream>>>(partials, (float*)d_out,
                                          LOSS_BLOCKS, 1.0f / (float)NPIX);
}